// GCNEncoder_77567109365801
// MI455X (gfx1250) — compile-verified
//
#include <hip/hip_runtime.h>

typedef __attribute__((ext_vector_type(2))) float v2f;
typedef __attribute__((ext_vector_type(8))) float v8f;

// ---------------------------------------------------------------------------
// Degree / normalization kernels
// ---------------------------------------------------------------------------

// deg[i] = 1  (self loop)
__global__ void deg_init_kernel(int* __restrict__ deg, int n) {
    int i = blockIdx.x * blockDim.x + threadIdx.x;
    if (i < n) deg[i] = 1;
}

// deg[dst[e]] += 1  (integer atomics -> deterministic counts)
__global__ void deg_accum_kernel(const int* __restrict__ dst, int* __restrict__ deg, int E) {
    int e = blockIdx.x * blockDim.x + threadIdx.x;
    if (e < E) atomicAdd(&deg[dst[e]], 1);
}

// in-place: dinv[i] = rsqrt((float)deg[i])   (deg >= 1 always, no zero check)
__global__ void dinv_kernel(float* __restrict__ dinv, const int* __restrict__ deg, int n) {
    int i = blockIdx.x * blockDim.x + threadIdx.x;
    if (i < n) {
        float d = (float)deg[i];
        dinv[i] = rsqrtf(d);
    }
}

// norm[e] = dinv[src[e]] * dinv[dst[e]]
__global__ void norm_kernel(const int* __restrict__ src, const int* __restrict__ dst,
                            const float* __restrict__ dinv, float* __restrict__ norm, int E) {
    int e = blockIdx.x * blockDim.x + threadIdx.x;
    if (e < E) norm[e] = dinv[src[e]] * dinv[dst[e]];
}

// ---------------------------------------------------------------------------
// Full-precision WMMA GEMM:  C[nrows x NCOL] = A[nrows x K] * B[K x NCOL]
// One wave computes a 16-row strip across all NCOL columns using
// V_WMMA_F32_16X16X4_F32 (A fragment loaded once per k-block, reused
// across all column tiles).
// ---------------------------------------------------------------------------
template <int K, int NCOL>
__global__ __launch_bounds__(128) void gemm_wmma_f32(const float* __restrict__ A,
                                                     const float* __restrict__ B,
                                                     float* __restrict__ C, int nrows) {
    const int wave = threadIdx.x >> 5;
    const int lane = threadIdx.x & 31;
    const int half = lane >> 4;   // 0: K pair {0,1} / rows M0..7 of C ; 1: K pair {2,3} / rows M8..15
    const int mr   = lane & 15;   // A row within tile / C column within tile

    const int row_tile = blockIdx.x * (blockDim.x >> 5) + wave;
    const int row0 = row_tile * 16;
    if (row0 >= nrows) return;    // wave-uniform: EXEC stays all-ones for WMMA

    constexpr int NT = NCOL / 16;
    v8f acc[NT] = {};

    // A: lane holds A[row0+mr][kb*4 + half*2 + {0,1}]
    const float* arow = A + (size_t)(row0 + mr) * K + half * 2;

#pragma unroll 8
    for (int kb = 0; kb < K / 4; ++kb) {
        v2f a;
        a.x = arow[kb * 4 + 0];
        a.y = arow[kb * 4 + 1];
        // B: lane holds B[kb*4 + half*2 + {0,1}][ct*16 + mr]
        const float* brow = B + (size_t)(kb * 4 + half * 2) * NCOL + mr;
#pragma unroll
        for (int ct = 0; ct < NT; ++ct) {
            v2f b;
            b.x = brow[ct * 16];
            b.y = brow[ct * 16 + NCOL];
            acc[ct] = __builtin_amdgcn_wmma_f32_16x16x4_f32(
                /*neg_a=*/false, a, /*neg_b=*/false, b,
                /*c_mod=*/(short)0, acc[ct], /*reuse_a=*/false, /*reuse_b=*/false);
        }
    }

    // C layout: VGPR r -> row (row0 + half*8 + r), col (ct*16 + mr)
#pragma unroll
    for (int ct = 0; ct < NT; ++ct) {
        float* crow = C + (size_t)(row0 + half * 8) * NCOL + ct * 16 + mr;
#pragma unroll
        for (int r = 0; r < 8; ++r) crow[(size_t)r * NCOL] = acc[ct][r];
    }
}

// ---------------------------------------------------------------------------
// out[i,f] = b[f] + dinv[i]^2 * h[i,f]    (bias + fused self-loop message)
// ---------------------------------------------------------------------------
template <int F>
__global__ void self_bias_init_kernel(const float* __restrict__ h, const float* __restrict__ dinv,
                                      const float* __restrict__ b, float* __restrict__ out,
                                      long total) {
    long idx = (long)blockIdx.x * blockDim.x + threadIdx.x;
    if (idx >= total) return;
    constexpr int SH = (F == 128) ? 7 : 6;
    int  f = (int)(idx & (F - 1));
    long i = idx >> SH;
    float di = dinv[i];
    out[idx] = b[f] + di * di * h[idx];
}

__global__ void relu_kernel(float* __restrict__ x, long total) {
    long idx = (long)blockIdx.x * blockDim.x + threadIdx.x;
    if (idx < total) x[idx] = fmaxf(x[idx], 0.0f);
}

// ---------------------------------------------------------------------------
// Edge scatter: one wave per edge.
//   out[dst[e], :] += norm[e] * h[src[e], :]
// F=128: each lane does a float4 gather + 4 f32 atomic adds.
// F=64 : each lane does a float2 gather + 2 f32 atomic adds.
// ---------------------------------------------------------------------------
template <int F>
__global__ void scatter_edges_kernel(const float* __restrict__ h, const int* __restrict__ src,
                                     const int* __restrict__ dst, const float* __restrict__ norm,
                                     float* __restrict__ out, int E) {
    unsigned gt = blockIdx.x * blockDim.x + threadIdx.x;
    int e = (int)(gt >> 5);
    if (e >= E) return;
    int lane = threadIdx.x & 31;

    int   s = src[e];
    int   d = dst[e];
    float w = norm[e];

    if constexpr (F == 128) {
        float4 v = *((const float4*)(h + (size_t)s * F) + lane);
        float* o = out + (size_t)d * F + lane * 4;
        atomicAdd(o + 0, v.x * w);
        atomicAdd(o + 1, v.y * w);
        atomicAdd(o + 2, v.z * w);
        atomicAdd(o + 3, v.w * w);
    } else {
        float2 v = *((const float2*)(h + (size_t)s * F) + lane);
        float* o = out + (size_t)d * F + lane * 2;
        atomicAdd(o + 0, v.x * w);
        atomicAdd(o + 1, v.y * w);
    }
}

// ---------------------------------------------------------------------------
// Launch
// ---------------------------------------------------------------------------
extern "C" void kernel_launch(void* const* d_in, const int* in_sizes, int n_in,
                              void* d_out, int out_size, void* d_ws, size_t ws_size,
                              hipStream_t stream) {
    const float* x  = (const float*)d_in[0];
    const int*   ei = (const int*)d_in[1];
    const float* W1 = (const float*)d_in[2];
    const float* b1 = (const float*)d_in[3];
    const float* W2 = (const float*)d_in[4];
    const float* b2 = (const float*)d_in[5];

    const int N = in_sizes[0] / 128;   // 100000
    const int E = in_sizes[1] / 2;     // 1600000
    const int* src = ei;
    const int* dst = ei + E;

    // workspace layout (floats): dinv[N] | norm[E] | h1[N*128] | agg1[N*128]
    float* dinv = (float*)d_ws;
    float* norm = dinv + N;
    float* h1   = norm + E;
    float* agg1 = h1 + (size_t)N * 128;
    float* h2   = h1;                  // reuse h1 buffer after layer-1 scatter
    float* out  = (float*)d_out;

    const int T = 256;

    // --- normalization: D^-1/2 (A+I) D^-1/2 weights --------------------------
    deg_init_kernel<<<(N + T - 1) / T, T, 0, stream>>>((int*)dinv, N);
    deg_accum_kernel<<<(E + T - 1) / T, T, 0, stream>>>(dst, (int*)dinv, E);
    dinv_kernel<<<(N + T - 1) / T, T, 0, stream>>>(dinv, (const int*)dinv, N);
    norm_kernel<<<(E + T - 1) / T, T, 0, stream>>>(src, dst, dinv, norm, E);

    const int row_tiles = (N + 15) / 16;       // 6250 (N divisible by 16)
    const int gemm_blocks = (row_tiles + 3) / 4;

    // --- layer 1: z = relu( Ahat @ (x@W1) + b1 ) -----------------------------
    gemm_wmma_f32<128, 128><<<gemm_blocks, 128, 0, stream>>>(x, W1, h1, N);
    long tot1 = (long)N * 128;
    self_bias_init_kernel<128><<<(int)((tot1 + T - 1) / T), T, 0, stream>>>(h1, dinv, b1, agg1, tot1);
    scatter_edges_kernel<128><<<(E + 7) / 8, T, 0, stream>>>(h1, src, dst, norm, agg1, E);
    relu_kernel<<<(int)((tot1 + T - 1) / T), T, 0, stream>>>(agg1, tot1);

    // --- layer 2: out = Ahat @ (z@W2) + b2 -----------------------------------
    gemm_wmma_f32<128, 64><<<gemm_blocks, 128, 0, stream>>>(agg1, W2, h2, N);
    long tot2 = (long)N * 64;
    self_bias_init_kernel<64><<<(int)((tot2 + T - 1) / T), T, 0, stream>>>(h2, dinv, b2, out, tot2);
    scatter_edges_kernel<64><<<(E + 7) / 8, T, 0, stream>>>(h2, src, dst, norm, out, E);
}